// Downsample1d_4827543241179
// MI455X (gfx1250) — compile-verified
//
#include <hip/hip_runtime.h>

// Depthwise 1D conv, K=4, stride=4, reflect pad 1, fp32.
// x: [8, 512, 16384] -> y: [8, 512, 4096].
// Memory-bound (320 MiB moved, ~0.35 FLOP/byte): optimized as a pure stream:
//   - b128 non-temporal loads (input read exactly once, > L2 capacity)
//   - 4 outputs per thread, b128 non-temporal store
//   - wave32 shuffle for the single halo element per thread

typedef float v4f __attribute__((ext_vector_type(4)));

#define T_IN  16384
#define T_OUT 4096
#define GROUPS_PER_ROW 1024   // T_OUT / 4 outputs per thread

__global__ __launch_bounds__(256) void downsample1d_kernel(
    const float* __restrict__ x,
    const float* __restrict__ kern,
    float* __restrict__ y,
    int n_groups)
{
    int gid = blockIdx.x * blockDim.x + threadIdx.x;
    if (gid >= n_groups) return;            // never taken: n_groups % 256 == 0

    int row = gid >> 10;                    // gid / GROUPS_PER_ROW
    int g   = gid & (GROUPS_PER_ROW - 1);   // group within row

    const float* xr = x + (size_t)row * T_IN + ((size_t)g << 4);

    // 64 B of streaming input per thread; TH=NT so the 256 MiB stream
    // doesn't thrash the 192 MB L2 (data is single-use).
    v4f v0 = __builtin_nontemporal_load((const v4f*)(xr + 0));
    v4f v1 = __builtin_nontemporal_load((const v4f*)(xr + 4));
    v4f v2 = __builtin_nontemporal_load((const v4f*)(xr + 8));
    v4f v3 = __builtin_nontemporal_load((const v4f*)(xr + 12));

    // Uniform kernel weights -> scalar loads (broadcast, cached).
    float k0 = kern[0], k1 = kern[1], k2 = kern[2], k3 = kern[3];

    // Halo: x[16*i - 1] is lane-1's v3.w. wave32 shuffle; lane 0 patches up.
    float prev = __shfl_up(v3.w, 1, 32);
    if ((threadIdx.x & 31) == 0) {
        // g is a multiple of 32 here (1024 groups/row, 32 lanes/wave),
        // so g == 0 <=> row start: reflect pad x[-1] -> x[1] = v0.y.
        prev = (g == 0) ? v0.y : xr[-1];
    }

    v4f o;
    o.x = k0*prev  + k1*v0.x + k2*v0.y + k3*v0.z;
    o.y = k0*v0.w + k1*v1.x + k2*v1.y + k3*v1.z;
    o.z = k0*v1.w + k1*v2.x + k2*v2.y + k3*v2.z;
    o.w = k0*v2.w + k1*v3.x + k2*v3.y + k3*v3.z;

    // Output is write-once, never re-read by this kernel -> NT store.
    __builtin_nontemporal_store(o, (v4f*)(y + (size_t)row * T_OUT + ((size_t)g << 2)));
}

extern "C" void kernel_launch(void* const* d_in, const int* in_sizes, int n_in,
                              void* d_out, int out_size, void* d_ws, size_t ws_size,
                              hipStream_t stream)
{
    const float* x    = (const float*)d_in[0];   // hidden_states, fp32
    const float* kern = (const float*)d_in[1];   // kernel [4], fp32
    float*       y    = (float*)d_out;           // fp32 output

    int rows     = in_sizes[0] / T_IN;           // B*C = 4096
    int n_groups = rows * GROUPS_PER_ROW;        // 4,194,304 threads

    const int block = 256;                       // 8 waves (wave32) per block
    int grid = (n_groups + block - 1) / block;

    downsample1d_kernel<<<grid, block, 0, stream>>>(x, kern, y, n_groups);
}